// ScaledDotProductAttention_67070209294956
// MI455X (gfx1250) — compile-verified
//
#include <hip/hip_runtime.h>
#include <math.h>

// Sinkhorn attention for MI455X (gfx1250, wave32).
// attn = sigmoid(clip(q.k^T/8)) kept in factored form s_ij * u_i * v_j;
// the 5 row/col normalizations reduce to 10 mat-vecs (u/v scaling vectors),
// so only the mandatory ~166 MB touches HBM; everything else is L2-resident
// (whole attn tensor = 134 MB < 192 MB L2).
// Matmuls use V_WMMA_F32_16X16X4_F32 (full fp32 fidelity; compute is not the
// bottleneck at 23.3 TB/s).

typedef __attribute__((ext_vector_type(2))) float v2f;
typedef __attribute__((ext_vector_type(8))) float v8f;

constexpr int B = 4, H = 8, S = 1024, D = 64;
constexpr int BH = B * H;
constexpr float TEMP_INV = 0.125f;   // 1/TEMPERATURE
constexpr float EPS = 1e-9f;
constexpr size_t SS = (size_t)S * S; // per-head attn elements
constexpr size_t SD = (size_t)S * D; // per-head q/k/v/out elements

// ---------------------------------------------------------------------------
// Kernel 1: s = sigmoid(clip(q.k^T / 8)) -> attn region (pre-normalization).
// One wave per 16x16 tile; 16 x V_WMMA_F32_16X16X4_F32 over D=64.
// f32 16x4 A layout (ISA 7.12.2): lanes 0-15 -> M=lane, K=0..1 in v0..v1;
// lanes 16-31 -> M=lane-16, K=2..3.  B (4x16) mirrors it (K rows across the
// two VGPRs / lane halves, N across lanes) -> one float2 load per operand.
// ---------------------------------------------------------------------------
__global__ __launch_bounds__(256)
void qk_sigmoid_kernel(const float* __restrict__ q,
                       const float* __restrict__ k,
                       float* __restrict__ attn) {
  const int lane = threadIdx.x & 31;
  const int wave = threadIdx.x >> 5;
  const int gw   = blockIdx.x * 8 + wave;   // global wave id
  const int head = gw >> 12;                // 4096 tiles per head (64x64)
  const int tile = gw & 4095;
  const int i0 = (tile >> 6) * 16;
  const int j0 = (tile & 63) * 16;

  const float* qh = q + (size_t)head * SD;
  const float* kh = k + (size_t)head * SD;

  const int m  = lane & 15;
  const int kk = (lane >> 4) * 2;           // K sub-pair within the 4-chunk

  v8f c = {0.f, 0.f, 0.f, 0.f, 0.f, 0.f, 0.f, 0.f};
#pragma unroll
  for (int d0 = 0; d0 < D; d0 += 4) {
    const float2 av = *(const float2*)(qh + (size_t)(i0 + m) * D + d0 + kk);
    const float2 bv = *(const float2*)(kh + (size_t)(j0 + m) * D + d0 + kk);
    v2f a; a.x = av.x; a.y = av.y;
    v2f b; b.x = bv.x; b.y = bv.y;
    c = __builtin_amdgcn_wmma_f32_16x16x4_f32(false, a, false, b,
                                              (short)0, c, false, false);
  }

  // C/D layout: VGPR r, lanes 0-15 -> row i0+r, lanes 16-31 -> row i0+8+r.
  float* aM = attn + (size_t)head * SS;
  const int rbase = i0 + 8 * (lane >> 4);
  const int col   = j0 + m;
#pragma unroll
  for (int r = 0; r < 8; ++r) {
    float x = c[r] * TEMP_INV;
    x = fminf(50.f, fmaxf(-50.f, x));
    aM[(size_t)(rbase + r) * S + col] = 1.f / (1.f + __expf(-x));
  }
}

// ---------------------------------------------------------------------------
// Kernel 2: Sinkhorn scaling vectors.  One workgroup per (b,h).
// a_ij = s_ij*u_i*v_j is invariant under row/col normalization:
//   row: u_i <- u_i / (u_i*(S v)_i + eps)
//   col: v_j <- v_j / (v_j*(S^T u)_j + eps)
// 5 iterations = 10 mat-vecs over the 4 MB s-matrix (L2-resident).
// ---------------------------------------------------------------------------
__global__ __launch_bounds__(256)
void sinkhorn_uv_kernel(const float* __restrict__ attn,
                        float* __restrict__ uv) {
  __shared__ float u_lds[S];
  __shared__ float v_lds[S];
  const int head = blockIdx.x;
  const float* sM = attn + (size_t)head * SS;
  const int t = threadIdx.x;
  const int lane = t & 31, wave = t >> 5;

  for (int i = t; i < S; i += 256) { u_lds[i] = 1.f; v_lds[i] = 1.f; }
  __syncthreads();

  for (int it = 0; it < 5; ++it) {
    // Row pass: wave per row, lanes stride columns (coalesced 128B/step).
    for (int i = wave; i < S; i += 8) {
      const float* row = sM + (size_t)i * S;
      float sum = 0.f;
      for (int j = lane; j < S; j += 32) sum += row[j] * v_lds[j];
#pragma unroll
      for (int off = 16; off; off >>= 1) sum += __shfl_xor(sum, off, 32);
      if (lane == 0) {
        const float ui = u_lds[i];
        u_lds[i] = ui / (ui * sum + EPS);
      }
    }
    __syncthreads();

    // Col pass: thread t owns columns t, t+256, t+512, t+768 (coalesced rows).
    float a0 = 0.f, a1 = 0.f, a2 = 0.f, a3 = 0.f;
    for (int i = 0; i < S; ++i) {
      const float ui = u_lds[i];               // LDS broadcast
      const float* row = sM + (size_t)i * S;
      a0 += row[t      ] * ui;
      a1 += row[t + 256] * ui;
      a2 += row[t + 512] * ui;
      a3 += row[t + 768] * ui;
    }
    float vj;
    vj = v_lds[t      ]; v_lds[t      ] = vj / (vj * a0 + EPS);
    vj = v_lds[t + 256]; v_lds[t + 256] = vj / (vj * a1 + EPS);
    vj = v_lds[t + 512]; v_lds[t + 512] = vj / (vj * a2 + EPS);
    vj = v_lds[t + 768]; v_lds[t + 768] = vj / (vj * a3 + EPS);
    __syncthreads();
  }

  float* uvh = uv + (size_t)head * 2 * S;
  for (int i = t; i < S; i += 256) {
    uvh[i]     = u_lds[i];
    uvh[S + i] = v_lds[i];
  }
}

// ---------------------------------------------------------------------------
// Kernel 3: attn_ij = s_ij * u_i * v_j, in place (float4 traffic).
// ---------------------------------------------------------------------------
__global__ __launch_bounds__(256)
void finalize_attn_kernel(float* __restrict__ attn,
                          const float* __restrict__ uv) {
  const size_t idx = (size_t)blockIdx.x * 256 + threadIdx.x; // float4 index
  constexpr size_t PER_HEAD = SS / 4;                         // 262144
  const int head = (int)(idx / PER_HEAD);
  const size_t e = idx - (size_t)head * PER_HEAD;
  const int i  = (int)(e >> 8);            // S/4 = 256 float4 per row
  const int j0 = (int)((e & 255) << 2);

  const float* uvh = uv + (size_t)head * 2 * S;
  const float  ui = uvh[i];
  const float4 vv = *(const float4*)(uvh + S + j0);

  float4* p = (float4*)(attn + (size_t)head * SS + (size_t)i * S + j0);
  float4 sv = *p;
  sv.x *= ui * vv.x; sv.y *= ui * vv.y;
  sv.z *= ui * vv.z; sv.w *= ui * vv.w;
  *p = sv;
}

// ---------------------------------------------------------------------------
// Kernel 4: out = attn . V  (1024x1024 @ 1024x64 per head).
// One wave per 16x16 output tile; 256 chained f32 WMMAs over K=1024.
// ---------------------------------------------------------------------------
__global__ __launch_bounds__(256)
void av_gemm_kernel(const float* __restrict__ attn,
                    const float* __restrict__ v,
                    float* __restrict__ out) {
  const int lane = threadIdx.x & 31;
  const int wave = threadIdx.x >> 5;
  const int gw   = blockIdx.x * 8 + wave;
  const int head = gw >> 8;                 // 256 tiles per head (64 i x 4 d)
  const int tile = gw & 255;
  const int i0 = (tile >> 2) * 16;
  const int d0 = (tile & 3) * 16;

  const float* aM = attn + (size_t)head * SS;
  const float* vM = v    + (size_t)head * SD;

  const int m  = lane & 15;
  const int kk = (lane >> 4) * 2;

  v8f c = {0.f, 0.f, 0.f, 0.f, 0.f, 0.f, 0.f, 0.f};
  for (int j0 = 0; j0 < S; j0 += 4) {
    const float2 av2 = *(const float2*)(aM + (size_t)(i0 + m) * S + j0 + kk);
    v2f a; a.x = av2.x; a.y = av2.y;
    v2f b;
    b.x = vM[(size_t)(j0 + kk    ) * D + d0 + m];   // B[K][N] = V[j0+K][d0+N]
    b.y = vM[(size_t)(j0 + kk + 1) * D + d0 + m];
    c = __builtin_amdgcn_wmma_f32_16x16x4_f32(false, a, false, b,
                                              (short)0, c, false, false);
  }

  float* oM = out + (size_t)head * SD;
  const int rbase = i0 + 8 * (lane >> 4);
#pragma unroll
  for (int r = 0; r < 8; ++r)
    oM[(size_t)(rbase + r) * D + d0 + m] = c[r];
}

// ---------------------------------------------------------------------------
extern "C" void kernel_launch(void* const* d_in, const int* in_sizes, int n_in,
                              void* d_out, int out_size, void* d_ws, size_t ws_size,
                              hipStream_t stream) {
  const float* q = (const float*)d_in[0];
  const float* k = (const float*)d_in[1];
  const float* v = (const float*)d_in[2];

  float* out  = (float*)d_out;                      // [B,H,S,D]
  float* attn = (float*)d_out + (size_t)BH * SD;    // [B,H,S,S] (tuple slot 2)
  float* uv   = (float*)d_ws;                       // [BH][2][S] scaling vectors

  // 1) s = sigmoid(clip(q.k^T/8)) : 32 heads * 4096 tiles / 8 waves per block
  qk_sigmoid_kernel<<<BH * 4096 / 8, 256, 0, stream>>>(q, k, attn);
  // 2) Sinkhorn scaling vectors u,v per head
  sinkhorn_uv_kernel<<<BH, 256, 0, stream>>>(attn, uv);
  // 3) attn = s * u_i * v_j (in place)
  finalize_attn_kernel<<<(unsigned)((BH * SS / 4) / 256), 256, 0, stream>>>(attn, uv);
  // 4) out = attn . V : 32 heads * 256 tiles / 8 waves per block
  av_gemm_kernel<<<BH * 256 / 8, 256, 0, stream>>>(attn, v, out);
}